// COIL_Core_6554120094109
// MI455X (gfx1250) — compile-verified
//
#include <hip/hip_runtime.h>
#include <hip/hip_bf16.h>

// ---------------------------------------------------------------------------
// B=32, S=2048, H=768, TOK=128, CLS=768, WINDOW=5, EPS=1e-5.  All fp32.
// Output = [cls_rep (32x768) | reps (32x2046x128)] flat.
//
// Workspace layout (floats):
//   reps_masked : 32*2046*128 = 8,380,416   @ 0
//   cls_lin     : 32*768      =    24,576   @ 8,380,416
//   wr_tok      : 4*192*32*4  =    98,304   @ 8,404,992
//   wr_cls      : 24*192*32*4 =   589,824   @ 8,503,296
//   lengths(int): 32                         @ 9,093,120
// total ~36.4 MB of d_ws.
// ---------------------------------------------------------------------------

typedef float v2f __attribute__((ext_vector_type(2)));
typedef float v8f __attribute__((ext_vector_type(8)));

__device__ __forceinline__ v8f wmma_f32(v2f a, v2f b, v8f c) {
    // V_WMMA_F32_16X16X4_F32:  D = A(16x4) * B(4x16) + C(16x16), all fp32.
    return __builtin_amdgcn_wmma_f32_16x16x4_f32(
        false, a, false, b, (short)0, c, false, false);
}

// ---------------------------------------------------------------------------
// Reorder a (768 x N) weight matrix into per-lane paired-tile B-fragment
// streams: one float4 per lane per K-step covers TWO 16-col tiles:
//   Wr[((pair*192 + i)*32 + lane)*4 + tl*2 + j]
//     = W[(4*i + 2*(lane>>4) + j) * N + (pair*2 + tl)*16 + (lane&15)]
// -> the GEMM inner loop does a single coalesced global_load_b128 per K-step.
// ---------------------------------------------------------------------------
__global__ __launch_bounds__(256) void reorder_w(const float* __restrict__ W,
                                                 int N,
                                                 float* __restrict__ Wr) {
    int o    = blockIdx.x * 256 + threadIdx.x;   // grid sized exactly
    int pair = o / 24576;                        // 192*32*4
    int rem  = o % 24576;
    int i    = rem / 128;
    int rem2 = rem % 128;
    int lane = rem2 >> 2;
    int tl   = (rem2 >> 1) & 1;
    int j    = rem2 & 1;
    int m    = lane & 15;
    int half = lane >> 4;
    int k    = 4 * i + 2 * half + j;
    int n    = (pair * 2 + tl) * 16 + m;
    Wr[o] = W[k * N + n];
}

// ---------------------------------------------------------------------------
// Per-batch valid length of the sliced mask: L[b] = min(sum(mask[b])-1, 2046)
// ---------------------------------------------------------------------------
__global__ __launch_bounds__(256) void lengths_k(const int* __restrict__ mask,
                                                 int* __restrict__ Lout) {
    __shared__ int sh[8];
    int b = blockIdx.x, t = threadIdx.x;
    int s = 0;
    #pragma unroll
    for (int j = 0; j < 8; ++j) s += mask[b * 2048 + t + 256 * j];
    #pragma unroll
    for (int o = 1; o < 32; o <<= 1) s += __shfl_xor(s, o, 32);
    if ((t & 31) == 0) sh[t >> 5] = s;
    __syncthreads();
    if (t == 0) {
        int tot = 0;
        #pragma unroll
        for (int j = 0; j < 8; ++j) tot += sh[j];
        int L = tot - 1;
        Lout[b] = L < 2046 ? L : 2046;
    }
}

// ---------------------------------------------------------------------------
// Kernel A: tok projection with WMMA f32 16x16x4, then +bias, LayerNorm(128),
// ReLU, mask; store masked reps for rows s in [1, 2046].
//
// Block = 128 threads (4 waves) = 64 rows.  Each wave: 4 M-tiles x 2 N-tiles
// (8 accumulators) -> each B fragment (one global_load_b128) feeds 4 WMMAs,
// cutting L2 weight traffic 4x vs a 1-M-tile layout.  K is processed in 4
// phases of 192 so the staged A tile (64 x 196 floats, 50 KB) fits LDS; the
// LayerNorm C buffer (64 x 132) aliases the same LDS.
// LDA=196 => dword index = 4m + 4i + 2*half (mod 64): conflict-free b64 reads.
// ---------------------------------------------------------------------------
#define LDA 196
#define LDC 132
#define KP  192   // K per phase
#define IT  48    // WMMA K-steps per phase

__global__ __launch_bounds__(128) void tok_gemm_ln(
    const float* __restrict__ hidden, const int* __restrict__ amask,
    const float* __restrict__ wr,     const float* __restrict__ tok_b,
    const float* __restrict__ gamma,  const float* __restrict__ beta,
    float* __restrict__ reps) {
    __shared__ __align__(16) float shmem[64 * LDA];   // 50,176 B
    float* shA = shmem;                                // phase-staged A tile
    float* shC = shmem;                                // aliased C tile (64x132)

    const int  t    = threadIdx.x;
    const int  lane = t & 31;
    const int  wv   = t >> 5;            // 0..3 -> N pair
    const int  m    = lane & 15;
    const int  half = lane >> 4;
    const long row0 = (long)blockIdx.x * 64;

    v8f acc[4][2] = {};                                // [m-tile][n-tile]
    const float4* bp = (const float4*)wr + ((long)wv * 192) * 32 + lane;

    for (int ph = 0; ph < 4; ++ph) {
        __syncthreads();                 // previous phase done reading shA
        // Stage 64 rows x 192 cols of hidden (coalesced float4).
        for (int idx = t; idx < 64 * KP / 4; idx += 128) {
            int r  = idx / (KP / 4);
            int c4 = (idx % (KP / 4)) * 4;
            float4 v = *(const float4*)(hidden + (row0 + r) * 768 + ph * KP + c4);
            *(float4*)&shA[r * LDA + c4] = v;
        }
        __syncthreads();

        for (int i = 0; i < IT; ++i) {
            float4 f = bp[(long)(ph * IT + i) * 32];   // global_load_b128
            v2f b0 = {f.x, f.y};
            v2f b1 = {f.z, f.w};
            #pragma unroll
            for (int mt = 0; mt < 4; ++mt) {
                v2f a = *(const v2f*)&shA[(mt * 16 + m) * LDA + 4 * i + 2 * half];
                acc[mt][0] = wmma_f32(a, b0, acc[mt][0]);
                acc[mt][1] = wmma_f32(a, b1, acc[mt][1]);
            }
        }
    }
    __syncthreads();                     // all waves done with shA before alias

    // Scatter C tiles (M=i and M=i+8 halves per ISA layout) into padded LDS.
    #pragma unroll
    for (int mt = 0; mt < 4; ++mt) {
        #pragma unroll
        for (int i = 0; i < 8; ++i) {
            shC[(mt * 16 + i + 8 * half) * LDC + wv * 32 + m]      = acc[mt][0][i];
            shC[(mt * 16 + i + 8 * half) * LDC + wv * 32 + 16 + m] = acc[mt][1][i];
        }
    }
    __syncthreads();

    // LayerNorm(128) + ReLU + mask: 2 lanes per row, 64 cols per lane.
    const int r  = t >> 1;               // 0..63
    const int c0 = (t & 1) * 64;
    float s = 0.f, ss = 0.f;
    for (int i = 0; i < 64; ++i) {
        float v = shC[r * LDC + c0 + i] + tok_b[c0 + i];
        s += v; ss += v * v;
    }
    s  += __shfl_xor(s,  1, 32);
    ss += __shfl_xor(ss, 1, 32);
    const float mean = s * (1.f / 128.f);
    const float var  = ss * (1.f / 128.f) - mean * mean;
    const float rstd = rsqrtf(var + 1e-5f);

    const long rg = row0 + r;
    const int  bI = (int)(rg >> 11);
    const int  sI = (int)(rg & 2047);
    const float mk = (float)amask[rg];
    if (sI >= 1 && sI <= 2046) {
        float* dst = reps + ((long)(bI * 2046 + sI - 1)) * 128 + c0;
        for (int i = 0; i < 64; ++i) {
            float v = shC[r * LDC + c0 + i] + tok_b[c0 + i];
            float y = (v - mean) * rstd * gamma[c0 + i] + beta[c0 + i];
            dst[i] = fmaxf(y, 0.f) * mk;
        }
    }
}

// ---------------------------------------------------------------------------
// Kernel B: cls projection (32x768, K=768), WMMA f32 16x16x4.
// 6 blocks x 4 waves; each wave: one N pair (2 tiles) x two 16-row M-tiles.
// ---------------------------------------------------------------------------
__global__ __launch_bounds__(128) void cls_gemm(const float* __restrict__ hidden,
                                                const float* __restrict__ wr2,
                                                float* __restrict__ cls_lin) {
    const int t = threadIdx.x, lane = t & 31, wv = t >> 5;
    const int m = lane & 15, half = lane >> 4;
    const int pair = blockIdx.x * 4 + wv;     // 0..23

    v8f acc[2][2] = {};
    const float4* bp = (const float4*)wr2 + ((long)pair * 192) * 32 + lane;
    const float* a0b = hidden + (long)m * (2048 * 768) + 2 * half;         // rows 0..15 (s=0)
    const float* a1b = hidden + (long)(m + 16) * (2048 * 768) + 2 * half;  // rows 16..31

    for (int i = 0; i < 192; ++i) {
        v2f a0 = *(const v2f*)(a0b + 4 * i);
        v2f a1 = *(const v2f*)(a1b + 4 * i);
        float4 f = bp[i * 32];
        v2f b0 = {f.x, f.y};
        v2f b1 = {f.z, f.w};
        acc[0][0] = wmma_f32(a0, b0, acc[0][0]);
        acc[0][1] = wmma_f32(a0, b1, acc[0][1]);
        acc[1][0] = wmma_f32(a1, b0, acc[1][0]);
        acc[1][1] = wmma_f32(a1, b1, acc[1][1]);
    }
    const int c0 = pair * 32 + m;
    #pragma unroll
    for (int i = 0; i < 8; ++i) {
        cls_lin[(i + 8 * half) * 768 + c0]           = acc[0][0][i];
        cls_lin[(i + 8 * half) * 768 + c0 + 16]      = acc[0][1][i];
        cls_lin[(i + 8 * half + 16) * 768 + c0]      = acc[1][0][i];
        cls_lin[(i + 8 * half + 16) * 768 + c0 + 16] = acc[1][1][i];
    }
}

// ---------------------------------------------------------------------------
// Kernel B2: cls bias + LayerNorm(768) -> d_out[0 : 32*768]
// ---------------------------------------------------------------------------
__global__ __launch_bounds__(256) void cls_ln(const float* __restrict__ cls_lin,
                                              const float* __restrict__ cls_b,
                                              const float* __restrict__ gamma,
                                              const float* __restrict__ beta,
                                              float* __restrict__ out) {
    __shared__ float sh_s[8], sh_ss[8];
    const int b = blockIdx.x, t = threadIdx.x;
    float x[3];
    float s = 0.f, ss = 0.f;
    #pragma unroll
    for (int j = 0; j < 3; ++j) {
        int c = t + 256 * j;
        float v = cls_lin[b * 768 + c] + cls_b[c];
        x[j] = v; s += v; ss += v * v;
    }
    #pragma unroll
    for (int o = 1; o < 32; o <<= 1) {
        s  += __shfl_xor(s,  o, 32);
        ss += __shfl_xor(ss, o, 32);
    }
    if ((t & 31) == 0) { sh_s[t >> 5] = s; sh_ss[t >> 5] = ss; }
    __syncthreads();
    float ts = 0.f, tss = 0.f;
    #pragma unroll
    for (int j = 0; j < 8; ++j) { ts += sh_s[j]; tss += sh_ss[j]; }
    const float mean = ts * (1.f / 768.f);
    const float var  = tss * (1.f / 768.f) - mean * mean;
    const float rstd = rsqrtf(var + 1e-5f);
    #pragma unroll
    for (int j = 0; j < 3; ++j) {
        int c = t + 256 * j;
        out[b * 768 + c] = (x[j] - mean) * rstd * gamma[c] + beta[c];
    }
}

// ---------------------------------------------------------------------------
// Kernel C: windowed masked mean (+-5, clamped by L) then L2-normalize.
// One wave per output row; float4 per lane.  8184 blocks * 8 == 32*2046.
// ---------------------------------------------------------------------------
__global__ __launch_bounds__(256) void window_norm(const float* __restrict__ reps,
                                                   const int* __restrict__ Lw,
                                                   float* __restrict__ out) {
    const int g    = blockIdx.x * 8 + (threadIdx.x >> 5);
    const int lane = threadIdx.x & 31;
    const int b    = g / 2046;
    const int tp   = g % 2046;
    const int L    = Lw[b];
    const bool valid = tp < L;
    int start = tp - 5; if (start < 0) start = 0;
    int end   = tp + 5; if (end > L) end = L;
    int cnt   = end - start; if (cnt < 1) cnt = 1;

    float4 sum = {0.f, 0.f, 0.f, 0.f};
    const float* base = reps + ((long)b * 2046) * 128 + lane * 4;
    for (int r = start; r < end; ++r) {
        float4 v = *(const float4*)(base + (long)r * 128);
        sum.x += v.x; sum.y += v.y; sum.z += v.z; sum.w += v.w;
    }
    const float inv = 1.f / (float)cnt;
    float4 mn = {sum.x * inv, sum.y * inv, sum.z * inv, sum.w * inv};
    float sq = mn.x * mn.x + mn.y * mn.y + mn.z * mn.z + mn.w * mn.w;
    #pragma unroll
    for (int o = 1; o < 32; o <<= 1) sq += __shfl_xor(sq, o, 32);
    const float n  = sqrtf(sq);
    const float sc = (valid && n > 0.f) ? (1.f / n) : 0.f;
    float4 o4 = {mn.x * sc, mn.y * sc, mn.z * sc, mn.w * sc};
    *(float4*)(out + (long)g * 128 + lane * 4) = o4;
}

// ---------------------------------------------------------------------------
extern "C" void kernel_launch(void* const* d_in, const int* in_sizes, int n_in,
                              void* d_out, int out_size, void* d_ws, size_t ws_size,
                              hipStream_t stream) {
    const float* hidden   = (const float*)d_in[0];
    const int*   amask    = (const int*)  d_in[1];
    const float* tok_w    = (const float*)d_in[2];
    const float* tok_b    = (const float*)d_in[3];
    const float* cls_w    = (const float*)d_in[4];
    const float* cls_b    = (const float*)d_in[5];
    const float* ln_tok_g = (const float*)d_in[6];
    const float* ln_tok_b = (const float*)d_in[7];
    const float* ln_cls_g = (const float*)d_in[8];
    const float* ln_cls_b = (const float*)d_in[9];
    float* out = (float*)d_out;

    float* ws      = (float*)d_ws;
    float* reps    = ws;                   // 8,380,416 floats
    float* cls_lin = ws + 8380416;         //    24,576
    float* wr_tok  = ws + 8404992;         //    98,304
    float* wr_cls  = ws + 8503296;         //   589,824
    int*   lenw    = (int*)(ws + 9093120); //        32

    reorder_w<<<384,  256, 0, stream>>>(tok_w, 128, wr_tok);
    reorder_w<<<2304, 256, 0, stream>>>(cls_w, 768, wr_cls);
    lengths_k<<<32,   256, 0, stream>>>(amask, lenw);

    tok_gemm_ln<<<1024, 128, 0, stream>>>(hidden, amask, wr_tok, tok_b,
                                          ln_tok_g, ln_tok_b, reps);
    cls_gemm<<<6, 128, 0, stream>>>(hidden, wr_cls, cls_lin);
    cls_ln<<<32, 256, 0, stream>>>(cls_lin, cls_b, ln_cls_g, ln_cls_b, out);
    window_norm<<<8184, 256, 0, stream>>>(reps, lenw, out + 24576);
}